// MSLoss_59811714564475
// MI455X (gfx1250) — compile-verified
//
#include <hip/hip_runtime.h>
#include <hip/hip_bf16.h>
#include <math.h>

typedef __attribute__((ext_vector_type(2))) float v2f;
typedef __attribute__((ext_vector_type(8))) float v8f;

#define NROWS 4096
#define KDIM  128
#define NT    256                 // 16-wide tiles per dimension
#define WAVES 16                  // waves per block
#define CHUNK_TILES (NT / WAVES)  // 16 column tiles per wave
#define NBLK  (NROWS / 32)        // 128 blocks: 32 rows (2 tiles) per block
#define ALPHA 10.0f
#define BETA  2.0f
#define BASEC 0.7f
#define EPSC  1e-6f
#define BIGC  1e9f

// monotonic float <-> uint mapping so atomicMin/atomicMax on uint == float min/max
__device__ __forceinline__ unsigned f2key(float f) {
    unsigned b = __float_as_uint(f);
    return (b & 0x80000000u) ? ~b : (b | 0x80000000u);
}
__device__ __forceinline__ float key2f(unsigned k) {
    unsigned b = (k & 0x80000000u) ? (k ^ 0x80000000u) : ~k;
    return __uint_as_float(b);
}

__global__ void k_init(unsigned* minkey, unsigned* maxkey, float* cntp, float* cntn,
                       float* sump, float* sumn, float* extras) {
    int i = blockIdx.x * blockDim.x + threadIdx.x;
    if (i < NROWS) {
        minkey[i] = 0xFFFFFFFFu;   // +max  -> min identity
        maxkey[i] = 0u;            // -max  -> max identity
        cntp[i] = 0.f; cntn[i] = 0.f; sump[i] = 0.f; sumn[i] = 0.f;
    }
    if (i < 8) extras[i] = 0.f;
}

// ---------------- Pass 1: per-row min positive sim / max negative sim ----------------
// 2x1 register blocking: each wave owns 32 rows (two A tiles), reuses each B k-step
// for two WMMAs into independent accumulators -> half the L2 B-traffic.
__global__ __launch_bounds__(WAVES * 32) void k_pass1(
    const float* __restrict__ x, const int* __restrict__ tg,
    unsigned* __restrict__ minkey, unsigned* __restrict__ maxkey) {

    const int lane  = threadIdx.x & 31;
    const int wave  = threadIdx.x >> 5;
    const int mBase = blockIdx.x * 32;
    const int l15   = lane & 15;
    const int half  = lane >> 4;          // 0 or 1
    const int kh    = half * 2;           // K sub-offset inside group of 4

    // Preload two A tiles (rows mBase..+15 and mBase+16..+31): 2*32 float2 = 128 VGPRs
    v2f A0[32], A1[32];
#pragma unroll
    for (int kk = 0; kk < 32; ++kk) {
        const float* p0 = x + (mBase + l15) * KDIM + kk * 4 + kh;
        const float* p1 = x + (mBase + 16 + l15) * KDIM + kk * 4 + kh;
        A0[kk].x = p0[0]; A0[kk].y = p0[1];
        A1[kk].x = p1[0]; A1[kk].y = p1[1];
    }
    int   trow[2][8];
    float minp[2][8], maxn[2][8];
#pragma unroll
    for (int t = 0; t < 2; ++t)
#pragma unroll
        for (int i = 0; i < 8; ++i) {
            trow[t][i] = tg[mBase + t * 16 + half * 8 + i];
            minp[t][i] = BIGC; maxn[t][i] = -BIGC;
        }

    const int colTile0 = wave * CHUNK_TILES;
    for (int ct = 0; ct < CHUNK_TILES; ++ct) {
        const int nBase = (colTile0 + ct) * 16;
        const int cCol  = nBase + l15;    // B/C: lane holds column N = lane%16
        v8f c0 = {0.f,0.f,0.f,0.f,0.f,0.f,0.f,0.f};
        v8f c1 = {0.f,0.f,0.f,0.f,0.f,0.f,0.f,0.f};
#pragma unroll
        for (int kk = 0; kk < 32; ++kk) {
            const float* p = x + cCol * KDIM + kk * 4 + kh;
            v2f b; b.x = p[0]; b.y = p[1];
            c0 = __builtin_amdgcn_wmma_f32_16x16x4_f32(false, A0[kk], false, b,
                                                       (short)0, c0, false, false);
            c1 = __builtin_amdgcn_wmma_f32_16x16x4_f32(false, A1[kk], false, b,
                                                       (short)0, c1, false, false);
        }
        const int tcol = tg[cCol];
#pragma unroll
        for (int t = 0; t < 2; ++t) {
            const v8f& c = t ? c1 : c0;
#pragma unroll
            for (int i = 0; i < 8; ++i) {
                const float s = c[i];
                const int   r = mBase + t * 16 + half * 8 + i;
                if (trow[t][i] == tcol) {
                    if (cCol != r && s < 1.0f - EPSC) minp[t][i] = fminf(minp[t][i], s);
                } else {
                    maxn[t][i] = fmaxf(maxn[t][i], s);
                }
            }
        }
    }
    // reduce each row value across the 16 lanes of its half-wave
#pragma unroll
    for (int t = 0; t < 2; ++t)
#pragma unroll
        for (int i = 0; i < 8; ++i)
#pragma unroll
            for (int m = 8; m >= 1; m >>= 1) {
                minp[t][i] = fminf(minp[t][i], __shfl_xor(minp[t][i], m));
                maxn[t][i] = fmaxf(maxn[t][i], __shfl_xor(maxn[t][i], m));
            }
    if (l15 < 8) {
#pragma unroll
        for (int t = 0; t < 2; ++t) {
            float mv = minp[t][0], xv = maxn[t][0];
#pragma unroll
            for (int j = 1; j < 8; ++j) if (l15 == j) { mv = minp[t][j]; xv = maxn[t][j]; }
            const int r = mBase + t * 16 + half * 8 + l15;
            atomicMin(&minkey[r], f2key(mv));
            atomicMax(&maxkey[r], f2key(xv));
        }
    }
}

// ---------------- Pass 2: mining masks + loss term accumulation ----------------
__global__ __launch_bounds__(WAVES * 32) void k_pass2(
    const float* __restrict__ x, const int* __restrict__ tg,
    const float* __restrict__ marg,
    const unsigned* __restrict__ minkey, const unsigned* __restrict__ maxkey,
    float* __restrict__ cntp, float* __restrict__ cntn,
    float* __restrict__ sump, float* __restrict__ sumn,
    float* __restrict__ extras) {

    const int lane  = threadIdx.x & 31;
    const int wave  = threadIdx.x >> 5;
    const int mBase = blockIdx.x * 32;
    const int l15   = lane & 15;
    const int half  = lane >> 4;
    const int kh    = half * 2;

    const float margin = marg[0];
    const float m10    = margin * 0.1f;
    const float fl10   = floorf(m10);

    v2f A0[32], A1[32];
#pragma unroll
    for (int kk = 0; kk < 32; ++kk) {
        const float* p0 = x + (mBase + l15) * KDIM + kk * 4 + kh;
        const float* p1 = x + (mBase + 16 + l15) * KDIM + kk * 4 + kh;
        A0[kk].x = p0[0]; A0[kk].y = p0[1];
        A1[kk].x = p1[0]; A1[kk].y = p1[1];
    }
    int   trow[2][8];
    float rowMin[2][8], rowMax[2][8];
    float cp[2][8], cn[2][8], sp[2][8], sn[2][8];
#pragma unroll
    for (int t = 0; t < 2; ++t)
#pragma unroll
        for (int i = 0; i < 8; ++i) {
            const int r = mBase + t * 16 + half * 8 + i;
            trow[t][i]   = tg[r];
            rowMin[t][i] = key2f(minkey[r]);
            rowMax[t][i] = key2f(maxkey[r]);
            cp[t][i] = cn[t][i] = sp[t][i] = sn[t][i] = 0.f;
        }
    float apCnt = 0.f;                                     // all pos pairs (length_AP)
    float lp_s = 0.f, lp_c = 0.f, ln_s = 0.f, ln_c = 0.f;  // last-row stats

    const int colTile0 = wave * CHUNK_TILES;
    for (int ct = 0; ct < CHUNK_TILES; ++ct) {
        const int nBase = (colTile0 + ct) * 16;
        const int cCol  = nBase + l15;
        v8f c0 = {0.f,0.f,0.f,0.f,0.f,0.f,0.f,0.f};
        v8f c1 = {0.f,0.f,0.f,0.f,0.f,0.f,0.f,0.f};
#pragma unroll
        for (int kk = 0; kk < 32; ++kk) {
            const float* p = x + cCol * KDIM + kk * 4 + kh;
            v2f b; b.x = p[0]; b.y = p[1];
            c0 = __builtin_amdgcn_wmma_f32_16x16x4_f32(false, A0[kk], false, b,
                                                       (short)0, c0, false, false);
            c1 = __builtin_amdgcn_wmma_f32_16x16x4_f32(false, A1[kk], false, b,
                                                       (short)0, c1, false, false);
        }
        const int tcol = tg[cCol];
#pragma unroll
        for (int t = 0; t < 2; ++t) {
            const v8f& c = t ? c1 : c0;
#pragma unroll
            for (int i = 0; i < 8; ++i) {
                const float s = c[i];
                const int   r = mBase + t * 16 + half * 8 + i;
                const bool same = (trow[t][i] == tcol);
                const bool pos  = same && (cCol != r) && (s < 1.0f - EPSC);
                const bool neg  = !same;
                if (pos) {
                    apCnt += 1.f;
                    if (rowMax[t][i] - s + margin > 0.f) {
                        cp[t][i] += 1.f;
                        sp[t][i] += log1pf(expf(-BETA * (s - BASEC)));
                    }
                }
                if (neg && (s + m10 - rowMin[t][i] > 0.f)) {
                    cn[t][i] += 1.f;
                    const float nv = s + m10 - fl10;
                    sn[t][i] += log1pf(expf(ALPHA * (nv - BASEC)));
                }
                if (r == NROWS - 1) {
                    if (pos) { lp_s += s; lp_c += 1.f; }
                    if (neg) { ln_s += s; ln_c += 1.f; }
                }
            }
        }
    }
    // per-row reductions across the 16 lanes of each half
#pragma unroll
    for (int t = 0; t < 2; ++t)
#pragma unroll
        for (int i = 0; i < 8; ++i)
#pragma unroll
            for (int m = 8; m >= 1; m >>= 1) {
                cp[t][i] += __shfl_xor(cp[t][i], m);
                cn[t][i] += __shfl_xor(cn[t][i], m);
                sp[t][i] += __shfl_xor(sp[t][i], m);
                sn[t][i] += __shfl_xor(sn[t][i], m);
            }
    if (l15 < 8) {
#pragma unroll
        for (int t = 0; t < 2; ++t) {
            float a0 = cp[t][0], a1 = cn[t][0], a2 = sp[t][0], a3 = sn[t][0];
#pragma unroll
            for (int j = 1; j < 8; ++j)
                if (l15 == j) { a0 = cp[t][j]; a1 = cn[t][j]; a2 = sp[t][j]; a3 = sn[t][j]; }
            const int r = mBase + t * 16 + half * 8 + l15;
            atomicAdd(&cntp[r], a0);
            atomicAdd(&cntn[r], a1);
            atomicAdd(&sump[r], a2);
            atomicAdd(&sumn[r], a3);
        }
    }
    // length_AP: full-wave reduction
#pragma unroll
    for (int m = 16; m >= 1; m >>= 1) apCnt += __shfl_xor(apCnt, m);
    if (lane == 0) atomicAdd(&extras[0], apCnt);
    // last-row pre-mining stats: nonzero only in half 1, tile 1, of the last block
#pragma unroll
    for (int m = 8; m >= 1; m >>= 1) {
        lp_s += __shfl_xor(lp_s, m); lp_c += __shfl_xor(lp_c, m);
        ln_s += __shfl_xor(ln_s, m); ln_c += __shfl_xor(ln_c, m);
    }
    if (lane == 16 && mBase + 31 == NROWS - 1) {
        atomicAdd(&extras[1], lp_s); atomicAdd(&extras[2], lp_c);
        atomicAdd(&extras[3], ln_s); atomicAdd(&extras[4], ln_c);
    }
}

// ---------------- Finalize: per-row losses -> 10 scalars ----------------
__global__ __launch_bounds__(256) void k_final(
    const float* __restrict__ cntp, const float* __restrict__ cntn,
    const float* __restrict__ sump, const float* __restrict__ sumn,
    const float* __restrict__ extras, float* __restrict__ out) {

    __shared__ float sh[4][256];
    const int tid = threadIdx.x;
    float accL = 0.f, accInv = 0.f, accAp = 0.f, accAn = 0.f;
    for (int r = tid; r < NROWS; r += 256) {
        const float cpv = cntp[r], cnv = cntn[r];
        const float pl = (2.0f / BETA)  * sump[r] / fmaxf(cpv, 1.0f);
        const float nl = (2.0f / ALPHA) * sumn[r] / fmaxf(cnv, 1.0f);
        const bool valid = (cpv > 0.f) && (cnv > 0.f);
        if (valid) { accL += pl + nl; accAp += cpv; accAn += cnv; }
        else       { accInv += 1.f; }
    }
    sh[0][tid] = accL; sh[1][tid] = accInv; sh[2][tid] = accAp; sh[3][tid] = accAn;
    __syncthreads();
    for (int st = 128; st > 0; st >>= 1) {
        if (tid < st) {
            sh[0][tid] += sh[0][tid + st]; sh[1][tid] += sh[1][tid + st];
            sh[2][tid] += sh[2][tid + st]; sh[3][tid] += sh[3][tid + st];
        }
        __syncthreads();
    }
    if (tid == 0) {
        const float nf   = (float)NROWS;
        const float loss = sh[0][0] / nf;
        const float prec = sh[1][0] / nf;
        const float lengthAP = extras[0];
        const float mp = extras[1] / fmaxf(extras[2], 1.f);
        const float mn = extras[3] / fmaxf(extras[4], 1.f);
        const float l_ap = sh[2][0] * 0.5f;
        const float l_an = sh[3][0] * 0.5f;
        const float r1 = l_an / l_ap;
        const float r2 = l_an / (lengthAP * 0.5f);
        out[0] = loss; out[1] = prec; out[2] = mp; out[3] = mn;
        out[4] = l_ap; out[5] = l_an;
        out[6] = r1; out[7] = 1.f / (1.f + expf(-r1));
        out[8] = r2; out[9] = 1.f / (1.f + expf(-r2));
    }
}

extern "C" void kernel_launch(void* const* d_in, const int* in_sizes, int n_in,
                              void* d_out, int out_size, void* d_ws, size_t ws_size,
                              hipStream_t stream) {
    const float* x    = (const float*)d_in[0];
    const int*   tg   = (const int*)d_in[1];
    const float* marg = (const float*)d_in[2];

    unsigned* minkey = (unsigned*)d_ws;
    unsigned* maxkey = minkey + NROWS;
    float* cntp   = (float*)(maxkey + NROWS);
    float* cntn   = cntp + NROWS;
    float* sump   = cntn + NROWS;
    float* sumn   = sump + NROWS;
    float* extras = sumn + NROWS;   // 8 floats
    float* out    = (float*)d_out;

    hipLaunchKernelGGL(k_init, dim3((NROWS + 255) / 256), dim3(256), 0, stream,
                       minkey, maxkey, cntp, cntn, sump, sumn, extras);
    hipLaunchKernelGGL(k_pass1, dim3(NBLK), dim3(WAVES * 32), 0, stream,
                       x, tg, minkey, maxkey);
    hipLaunchKernelGGL(k_pass2, dim3(NBLK), dim3(WAVES * 32), 0, stream,
                       x, tg, marg, minkey, maxkey, cntp, cntn, sump, sumn, extras);
    hipLaunchKernelGGL(k_final, dim3(1), dim3(256), 0, stream,
                       cntp, cntn, sump, sumn, extras, out);
}